// SAMGuidedLossBundle_8529805050144
// MI455X (gfx1250) — compile-verified
//
#include <hip/hip_runtime.h>
#include <hip/hip_bf16.h>

#define NSEG 17
#define NF   14
#define HH   512
#define WW   1024

typedef __attribute__((ext_vector_type(2))) float v2f;
typedef __attribute__((ext_vector_type(8))) float v8f;

// Per-feature operand codes: feature(m) = pick(acode[m]) * pick(bcode[m])
// code: 0->1.0, 1->x, 2->y, 3->u, 4->v, 5->0.0
// m:      0    1  2  3    4    5    6  7  8    9    10   11   12   13   14 15
// feat:   1    x  y  x*x  x*y  y*y  u  v  u*u  v*v  x*u  y*u  x*v  y*v  0  0
#define ACODE 0x5521214343211210ULL
#define BCODE 0x5544334300221000ULL

__device__ __forceinline__ float pick(int c, float x, float y, float u, float v) {
  float r = (c == 0) ? 1.0f : 0.0f;   // branchless v_cndmask chain
  r = (c == 1) ? x : r;
  r = (c == 2) ? y : r;
  r = (c == 3) ? u : r;
  r = (c == 4) ? v : r;
  return r;                            // c==5 stays 0
}

__device__ __forceinline__ float rdlane_f(int bits, int srclane) {
  return __int_as_float(__builtin_amdgcn_readlane(bits, srclane));
}

// ---------------- K0: zero workspace ----------------
__global__ void k_zero(float* __restrict__ ws, int n) {
  int i = blockIdx.x * blockDim.x + threadIdx.x;
  if (i < n) ws[i] = 0.0f;
}

// ---------------- K1: segment moments via WMMA f32 16x16x4 ----------------
// D[m][n] = sum_k feat(m, pixel_k) * onehot(seg_k == n+1): segment-sum as matmul.
// Columns map to segments 1..16 (background seg 0 is never consumed downstream).
// Broadcasts use v_readlane (constant lanes, no LDS, no DScnt waits).
__global__ void k_moments(const float* __restrict__ flow, const int* __restrict__ masks,
                          float* __restrict__ gmom, int blocksPerBatch) {
  __shared__ float lmom[NSEG * NF];
  constexpr int H = HH, W = WW, HW = H * W;
  const int b = blockIdx.x / blocksPerBatch;
  const int blkInB = blockIdx.x % blocksPerBatch;
  const int lane = threadIdx.x & 31;
  const int waveInBlk = threadIdx.x >> 5;
  const int wavesPerBlock = blockDim.x >> 5;
  const int waveInBatch = blkInB * wavesPerBlock + waveInBlk;
  const int wavesPerBatch = blocksPerBatch * wavesPerBlock;
  const int pixelsPerWave = HW / wavesPerBatch;   // 512: one row-aligned chunk per wave
  const int base = waveInBatch * pixelsPerWave;
  const float yf = (float)(base / W);             // whole chunk sits in one row
  const int x0 = base % W;

  for (int t = threadIdx.x; t < NSEG * NF; t += blockDim.x) lmom[t] = 0.0f;
  __syncthreads();

  const float* up = flow + (size_t)b * 2 * HW;
  const float* vp = up + HW;
  const int*   mp = masks + (size_t)b * HW;

  v8f acc = {};                          // D tile: features x segments(1..16)
  const int m  = lane & 15;              // A row (feature)
  const int np1 = (lane & 15) + 1;       // D column n -> segment n+1
  const int kh = (lane >= 16) ? 2 : 0;   // K half of A/B VGPR layout
  const bool lo = (lane < 16);
  const int ac = (int)((ACODE >> (4 * m)) & 0xFULL);  // lane-constant operand codes
  const int bc = (int)((BCODE >> (4 * m)) & 0xFULL);

  for (int it = 0; it < pixelsPerWave; it += 32) {
    const int idx = base + it + lane;
    const float u  = up[idx];
    const float v  = vp[idx];
    const int   sg = mp[idx];
    const int ub = __float_as_int(u);
    const int vb = __float_as_int(v);

#pragma unroll
    for (int g = 0; g < 8; ++g) {
      const int g4 = g * 4;
      // Broadcast the group's 4 pixels via readlane (constant source lanes),
      // then one cndmask selects the half-wave's pair (k=kh, kh+1).
      const float uA0 = rdlane_f(ub, g4 + 0), uA1 = rdlane_f(ub, g4 + 1);
      const float uB0 = rdlane_f(ub, g4 + 2), uB1 = rdlane_f(ub, g4 + 3);
      const float vA0 = rdlane_f(vb, g4 + 0), vA1 = rdlane_f(vb, g4 + 1);
      const float vB0 = rdlane_f(vb, g4 + 2), vB1 = rdlane_f(vb, g4 + 3);
      const int   sA0 = __builtin_amdgcn_readlane(sg, g4 + 0);
      const int   sA1 = __builtin_amdgcn_readlane(sg, g4 + 1);
      const int   sB0 = __builtin_amdgcn_readlane(sg, g4 + 2);
      const int   sB1 = __builtin_amdgcn_readlane(sg, g4 + 3);
      const float u0 = lo ? uA0 : uB0, u1 = lo ? uA1 : uB1;
      const float v0 = lo ? vA0 : vB0, v1 = lo ? vA1 : vB1;
      const int   s0 = lo ? sA0 : sB0, s1 = lo ? sA1 : sB1;
      const float xk0 = (float)(x0 + it + g4 + kh);   // x needs no broadcast
      const float xk1 = xk0 + 1.0f;
      v2f av;
      av.x = pick(ac, xk0, yf, u0, v0) * pick(bc, xk0, yf, u0, v0);
      av.y = pick(ac, xk1, yf, u1, v1) * pick(bc, xk1, yf, u1, v1);
      v2f bv;
      bv.x = (s0 == np1) ? 1.0f : 0.0f;
      bv.y = (s1 == np1) ? 1.0f : 0.0f;
      acc = __builtin_amdgcn_wmma_f32_16x16x4_f32(false, av, false, bv, (short)0, acc, false, false);
    }
  }

  // Flush the 16x16 D tile: lane holds D[r + hi*8][n] in acc[r], column n -> segment n+1.
#pragma unroll
  for (int r = 0; r < 8; ++r) {
    const int mm = r + ((lane >= 16) ? 8 : 0);
    if (mm < NF) atomicAdd(&lmom[np1 * NF + mm], acc[r]);
  }
  __syncthreads();
  for (int t = threadIdx.x; t < NSEG * NF; t += blockDim.x)
    atomicAdd(&gmom[(size_t)b * NSEG * NF + t], lmom[t]);
}

// ---------------- K2: 3x3 normal-equation solves + variance reduction ----------------
__global__ void k_solve(const float* __restrict__ gmom, float* __restrict__ params,
                        float* __restrict__ scal, int NS) {
  __shared__ float red[256];
  __shared__ float redc[256];
  const int t = threadIdx.x;
  float contrib = 0.0f, cnt = 0.0f;
  if (t < NS) {
    const float* mm = gmom + (size_t)t * NF;
    const float n = mm[0], Sx = mm[1], Sy = mm[2], Sxx = mm[3], Sxy = mm[4], Syy = mm[5];
    const float Su = mm[6], Sv = mm[7], Suu = mm[8], Svv = mm[9];
    const float Sxu = mm[10], Syu = mm[11], Sxv = mm[12], Syv = mm[13];
    const int segid = t % NSEG;
    const bool vh = (segid != 0) && (n >= 100.0f);
    float p[6] = {0, 0, 0, 0, 0, 0};
    if (vh) {
      const float a00 = Sxx, a01 = Sxy, a02 = Sx, a11 = Syy, a12 = Sy, a22 = n;
      const float c00 = a11 * a22 - a12 * a12;
      const float c01 = a02 * a12 - a01 * a22;
      const float c02 = a01 * a12 - a02 * a11;
      const float c11 = a00 * a22 - a02 * a02;
      const float c12 = a01 * a02 - a00 * a12;
      const float c22 = a00 * a11 - a01 * a01;
      const float det = a00 * c00 + a01 * c01 + a02 * c02;
      const float inv = (det != 0.0f) ? 1.0f / det : 0.0f;
      p[0] = inv * (c00 * Sxu + c01 * Syu + c02 * Su);
      p[1] = inv * (c01 * Sxu + c11 * Syu + c12 * Su);
      p[2] = inv * (c02 * Sxu + c12 * Syu + c22 * Su);
      p[3] = inv * (c00 * Sxv + c01 * Syv + c02 * Sv);
      p[4] = inv * (c01 * Sxv + c11 * Syv + c12 * Sv);
      p[5] = inv * (c02 * Sxv + c12 * Syv + c22 * Sv);
    }
    for (int j = 0; j < 6; ++j) params[t * 6 + j] = p[j];
    const bool vv = (segid != 0) && (n >= 50.0f);
    if (vv) {
      const float ns = fmaxf(n, 2.0f);
      const float var_u = (Suu - Su * Su / ns) / (ns - 1.0f);
      const float var_v = (Svv - Sv * Sv / ns) / (ns - 1.0f);
      contrib = var_u + var_v;
      cnt = 1.0f;
    }
  }
  red[t] = contrib; redc[t] = cnt;
  __syncthreads();
  for (int s2 = blockDim.x / 2; s2 > 0; s2 >>= 1) {
    if (t < s2) { red[t] += red[t + s2]; redc[t] += redc[t + s2]; }
    __syncthreads();
  }
  if (t == 0) { scal[0] = red[0]; scal[1] = redc[0]; }
}

// ---------------- helpers for K3/K4 ----------------
__device__ __forceinline__ float bndry(const int* __restrict__ mp, int y, int x) {
  constexpr int H = HH, W = WW;
  int mn = 0x7fffffff, mx = -0x7fffffff;
#pragma unroll
  for (int dy = -1; dy <= 1; ++dy)
#pragma unroll
    for (int dx = -1; dx <= 1; ++dx) {
      const int yy = y + dy, xx = x + dx;
      if (yy >= 0 && yy < H && xx >= 0 && xx < W) {
        const int s = mp[yy * W + xx];
        mn = min(mn, s); mx = max(mx, s);
      }
    }
  return (mn != mx) ? 1.0f : 0.0f;
}

__device__ __forceinline__ float magat(const float* __restrict__ up, const float* __restrict__ vp,
                                       int y, int x) {
  constexpr int H = HH, W = WW;
  if (y < 0 || y >= H || x < 0 || x >= W) return 0.0f;  // conv zero padding
  const float u = up[y * W + x], v = vp[y * W + x];
  return sqrtf(u * u + v * v);
}

__device__ __forceinline__ float fgat(const float* __restrict__ up, const float* __restrict__ vp,
                                      int y, int x) {
  const float m00 = magat(up, vp, y - 1, x - 1), m01 = magat(up, vp, y - 1, x),
              m02 = magat(up, vp, y - 1, x + 1);
  const float m10 = magat(up, vp, y, x - 1), m12 = magat(up, vp, y, x + 1);
  const float m20 = magat(up, vp, y + 1, x - 1), m21 = magat(up, vp, y + 1, x),
              m22 = magat(up, vp, y + 1, x + 1);
  const float gx = (m02 - m00) + 2.0f * (m12 - m10) + (m22 - m20);
  const float gy = (m20 - m00) + 2.0f * (m21 - m01) + (m22 - m02);
  return fabsf(gx) + fabsf(gy);
}

// ---------------- K3: residuals + boundary/fg maxes + smoothness ----------------
__global__ void k_main(const float* __restrict__ flow, const int* __restrict__ masks,
                       const float* __restrict__ images, const float* __restrict__ params,
                       float* __restrict__ res_sum, float* __restrict__ scal,
                       int blocksPerBatch) {
  __shared__ float lres[NSEG];
  __shared__ float redx[256], redy[256];
  constexpr int H = HH, W = WW, HW = H * W;
  const int b = blockIdx.x / blocksPerBatch;
  const int blkInB = blockIdx.x % blocksPerBatch;
  const float* up = flow + (size_t)b * 2 * HW;
  const float* vp = up + HW;
  const int*   mp = masks + (size_t)b * HW;
  const float* ip = images + (size_t)b * 3 * HW;
  for (int t = threadIdx.x; t < NSEG; t += blockDim.x) lres[t] = 0.0f;
  __syncthreads();

  float sumx = 0.0f, sumy = 0.0f, maxfg = 0.0f;
  int maxb = 0;
  const int stride = blocksPerBatch * blockDim.x;
  for (int idx = blkInB * blockDim.x + threadIdx.x; idx < HW; idx += stride) {
    __builtin_prefetch(up + idx + stride, 0, 0);   // speculative: dropped if invalid
    __builtin_prefetch(vp + idx + stride, 0, 0);
    __builtin_prefetch(mp + idx + stride, 0, 0);
    const int y = idx >> 10;            // W = 1024
    const int x = idx & (W - 1);
    const float u = up[idx], v = vp[idx];
    const int sg = mp[idx];
    // affine residual
    const float* p = params + ((size_t)b * NSEG + sg) * 6;
    const float fx = (float)x, fy = (float)y;
    const float puv = fx * p[0] + fy * p[1] + p[2];
    const float pvv = fx * p[3] + fy * p[4] + p[5];
    const float du = u - puv, dv = v - pvv;
    atomicAdd(&lres[sg], sqrtf(du * du + dv * dv));
    // boundary + sobel maxes
    const float bd = bndry(mp, y, x);
    if (bd > 0.0f) maxb = 1;
    maxfg = fmaxf(maxfg, fgat(up, vp, y, x));
    // boundary-aware smoothness
    const float g0 = (ip[idx] + ip[HW + idx] + ip[2 * HW + idx]) * (1.0f / 3.0f);
    if (x < W - 1) {
      const float dx0 = up[idx + 1] - u, dx1 = vp[idx + 1] - v;
      const float g1 = (ip[idx + 1] + ip[HW + idx + 1] + ip[2 * HW + idx + 1]) * (1.0f / 3.0f);
      float wx = expf(-fabsf(g1 - g0) * 10.0f);
      const int xi = (int)(((long long)x * W) / (W - 1));  // nearest-resize gather
      wx *= (1.0f - 0.9f * bndry(mp, y, xi));
      sumx += (fabsf(dx0) + fabsf(dx1)) * wx;
    }
    if (y < H - 1) {
      const float dy0 = up[idx + W] - u, dy1 = vp[idx + W] - v;
      const float g1 = (ip[idx + W] + ip[HW + idx + W] + ip[2 * HW + idx + W]) * (1.0f / 3.0f);
      float wy = expf(-fabsf(g1 - g0) * 10.0f);
      const int yi = (int)(((long long)y * H) / (H - 1));
      wy *= (1.0f - 0.9f * bndry(mp, yi, x));
      sumy += (fabsf(dy0) + fabsf(dy1)) * wy;
    }
  }
  redx[threadIdx.x] = sumx; redy[threadIdx.x] = sumy;
  __syncthreads();
  for (int s2 = blockDim.x / 2; s2 > 0; s2 >>= 1) {
    if (threadIdx.x < s2) {
      redx[threadIdx.x] += redx[threadIdx.x + s2];
      redy[threadIdx.x] += redy[threadIdx.x + s2];
    }
    __syncthreads();
  }
  if (threadIdx.x == 0) { atomicAdd(&scal[4], redx[0]); atomicAdd(&scal[5], redy[0]); }
  atomicMax((int*)&scal[2], __float_as_int(maxfg));  // fg >= 0: bit order == float order
  atomicMax((int*)&scal[3], maxb);
  for (int t = threadIdx.x; t < NSEG; t += blockDim.x)
    atomicAdd(&res_sum[b * NSEG + t], lres[t]);
}

// ---------------- K4: boundary sharpness (needs global maxes) ----------------
__global__ void k_sharp(const float* __restrict__ flow, const int* __restrict__ masks,
                        float* __restrict__ scal, int B) {
  __shared__ float red[256];
  constexpr int H = HH, W = WW, HW = H * W;
  const float maxfg = __int_as_float(((const int*)scal)[2]);
  const float maxbf = (float)((const int*)scal)[3];
  const long long total = (long long)B * HW;
  float acc = 0.0f;
  const long long gstride = (long long)gridDim.x * blockDim.x;
  for (long long i = (long long)blockIdx.x * blockDim.x + threadIdx.x; i < total; i += gstride) {
    const int b = (int)(i >> 19);       // HW = 2^19
    const int r = (int)(i & (HW - 1));
    const int y = r >> 10, x = r & (W - 1);
    const float* up = flow + (size_t)b * 2 * HW;
    const float* vp = up + HW;
    const int*   mp = masks + (size_t)b * HW;
    const float bn  = bndry(mp, y, x) / (maxbf + 1e-6f);
    const float fgn = fgat(up, vp, y, x) / (maxfg + 1e-6f);
    acc += bn * (1.0f - fgn);
  }
  red[threadIdx.x] = acc;
  __syncthreads();
  for (int s2 = blockDim.x / 2; s2 > 0; s2 >>= 1) {
    if (threadIdx.x < s2) red[threadIdx.x] += red[threadIdx.x + s2];
    __syncthreads();
  }
  if (threadIdx.x == 0) atomicAdd(&scal[6], red[0]);
}

// ---------------- K5: finalize ----------------
__global__ void k_final(const float* __restrict__ gmom, const float* __restrict__ res_sum,
                        const float* __restrict__ scal, float* __restrict__ out, int B) {
  constexpr int H = HH, W = WW;
  if (threadIdx.x != 0 || blockIdx.x != 0) return;
  float hs = 0.0f; int nh = 0;
  for (int s = 0; s < B * NSEG; ++s) {
    const float n = gmom[(size_t)s * NF];
    if ((s % NSEG) != 0 && n >= 100.0f) {
      hs += res_sum[s] / fmaxf(n, 1.0f);
      nh++;
    }
  }
  const float homog = hs / (float)max(nh, 1);
  const float ovar = scal[0] / fmaxf(scal[1], 1.0f);
  const float sharp = scal[6] / (float)((long long)B * H * W);
  const float smooth = scal[4] / (float)((long long)B * 2 * H * (W - 1)) +
                       scal[5] / (float)((long long)B * 2 * (H - 1) * W);
  out[0] = homog; out[1] = sharp; out[2] = ovar; out[3] = smooth;
}

extern "C" void kernel_launch(void* const* d_in, const int* in_sizes, int n_in,
                              void* d_out, int out_size, void* d_ws, size_t ws_size,
                              hipStream_t stream) {
  const float* flow   = (const float*)d_in[0];
  const int*   masks  = (const int*)d_in[1];
  const float* images = (const float*)d_in[2];
  float* out = (float*)d_out;

  const int B = in_sizes[1] / (HH * WW);
  const int NS = B * NSEG;

  float* ws = (float*)d_ws;
  float* gmom    = ws;                      // NS*NF
  float* params  = gmom + NS * NF;          // NS*6
  float* res_sum = params + NS * 6;         // NS
  float* scal    = res_sum + NS;            // 8 scalars
  const int nzero = NS * NF + NS * 6 + NS + 8;

  k_zero<<<(nzero + 255) / 256, 256, 0, stream>>>(ws, nzero);

  const int bpbMom = 128;   // 1024 waves/batch -> 512 px/wave, row-aligned chunks
  k_moments<<<B * bpbMom, 256, 0, stream>>>(flow, masks, gmom, bpbMom);

  k_solve<<<1, 256, 0, stream>>>(gmom, params, scal, NS);

  const int bpbMain = 128;
  k_main<<<B * bpbMain, 256, 0, stream>>>(flow, masks, images, params, res_sum, scal, bpbMain);

  k_sharp<<<2048, 256, 0, stream>>>(flow, masks, scal, B);

  k_final<<<1, 1, 0, stream>>>(gmom, res_sum, scal, out, B);
}